// EntropicGCN_76063870812586
// MI455X (gfx1250) — compile-verified
//
#include <hip/hip_runtime.h>
#include <math.h>

typedef float v2f __attribute__((ext_vector_type(2)));
typedef float v8f __attribute__((ext_vector_type(8)));

#define DH      128
#define TEMP_F  10.0f
#define LN_EPS  1e-5f
#define EPB     4        // edges per block in scatter kernels

// ---------------------------------------------------------------------------
// GEMM: H[M,Nc] = X[M,K] @ W[K,Nc], fp32, one wave per 16x16 output tile,
// K-loop of V_WMMA_F32_16X16X4_F32. Requires M%16==0, K%4==0, Nc%16==0.
// ---------------------------------------------------------------------------
__global__ __launch_bounds__(32)
void gemm_wmma_f32(const float* __restrict__ X, const float* __restrict__ W,
                   float* __restrict__ H, int M, int K, int Nc) {
  const int lane = threadIdx.x;
  const int r    = lane & 15;   // row of A / col of B,D
  const int half = lane >> 4;   // K sub-pair select (A/B), row-half select (D)
  const int kb   = half * 2;
  const int m0   = blockIdx.x * 16;
  const int n0   = blockIdx.y * 16;
  (void)M;
  v8f acc = {};
  for (int k = 0; k < K; k += 4) {
    v2f a, b;
    // A 16x4: VGPR0 = K{0,2}, VGPR1 = K{1,3} split by lane half
    a.x = X[(m0 + r) * K + (k + kb)];
    a.y = X[(m0 + r) * K + (k + kb + 1)];
    // B 4x16: same striping over N
    b.x = W[(k + kb) * Nc + (n0 + r)];
    b.y = W[(k + kb + 1) * Nc + (n0 + r)];
    acc = __builtin_amdgcn_wmma_f32_16x16x4_f32(
        /*neg_a=*/false, a, /*neg_b=*/false, b,
        /*c_mod=*/(short)0, acc, /*reuse_a=*/false, /*reuse_b=*/false);
  }
  // D 16x16: VGPR i = row i (lanes 0-15) / row 8+i (lanes 16-31)
  for (int i = 0; i < 8; ++i)
    H[(m0 + half * 8 + i) * Nc + (n0 + r)] = acc[i];
}

// ---------------------------------------------------------------------------
// Degrees: degIn[dst]++ (GCN, +1 self loop later), degO[src]++ (A row sums)
// ---------------------------------------------------------------------------
__global__ void degree_kernel(const int* __restrict__ src, const int* __restrict__ dst,
                              int E, float* degIn, float* degO) {
  int e = blockIdx.x * blockDim.x + threadIdx.x;
  if (e < E) {
    atomicAdd(&degIn[dst[e]], 1.0f);
    atomicAdd(&degO[src[e]], 1.0f);
  }
}

__global__ void dinv_kernel(const float* __restrict__ degIn, float* dinv, int N) {
  int i = blockIdx.x * blockDim.x + threadIdx.x;
  if (i < N) dinv[i] = rsqrtf(fmaxf(degIn[i] + 1.0f, 1e-12f));
}

// ---------------------------------------------------------------------------
// GCN aggregate: g = selfloop(h)*dinv^2 + bias, then edge scatter
// ---------------------------------------------------------------------------
__global__ void agg_init_kernel(float* __restrict__ g, const float* __restrict__ h,
                                const float* __restrict__ dinv,
                                const float* __restrict__ bias, int N, int Nc) {
  int idx = blockIdx.x * blockDim.x + threadIdx.x;
  if (idx < N * Nc) {
    int i = idx / Nc, f = idx % Nc;
    float di = dinv[i];
    g[idx] = h[idx] * di * di + bias[f];
  }
}

__global__ void agg_edge_kernel(float* __restrict__ g, const float* __restrict__ h,
                                const int* __restrict__ src, const int* __restrict__ dst,
                                const float* __restrict__ dinv, int E, int Nc) {
  int t  = threadIdx.x;
  int e0 = blockIdx.x * EPB;
  int e1 = min(e0 + EPB, E);
  for (int e = e0; e < e1; ++e) {
    int s = src[e], d = dst[e];
    float nrm = dinv[s] * dinv[d];
    atomicAdd(&g[d * Nc + t], h[s * Nc + t] * nrm);
  }
}

// ---------------------------------------------------------------------------
// Entropy-gradient pieces
// ---------------------------------------------------------------------------
__global__ __launch_bounds__(DH)
void node_sq_kernel(float* __restrict__ sq, const float* __restrict__ g) {
  __shared__ float sh[DH];
  int i = blockIdx.x, t = threadIdx.x;
  float v = g[i * DH + t];
  sh[t] = v * v; __syncthreads();
  for (int o = DH / 2; o > 0; o >>= 1) { if (t < o) sh[t] += sh[t + o]; __syncthreads(); }
  if (t == 0) sq[i] = sh[0];
}

// Ax[src] += g[dst]; Asq[src] += sq[dst]
__global__ void ax_scatter_kernel(float* __restrict__ Ax, float* __restrict__ Asq,
                                  const float* __restrict__ g, const float* __restrict__ sq,
                                  const int* __restrict__ src, const int* __restrict__ dst, int E) {
  int t  = threadIdx.x;
  int e0 = blockIdx.x * EPB;
  int e1 = min(e0 + EPB, E);
  for (int e = e0; e < e1; ++e) {
    int s = src[e], d = dst[e];
    atomicAdd(&Ax[s * DH + t], g[d * DH + t]);
    if (t == 0) atomicAdd(&Asq[s], sq[d]);
  }
}

// En = 0.5*(degO*sq + Asq - 2*sum(g*Ax))
__global__ __launch_bounds__(DH)
void en_kernel(float* __restrict__ En, const float* __restrict__ g,
               const float* __restrict__ Ax, const float* __restrict__ sq,
               const float* __restrict__ Asq, const float* __restrict__ degO) {
  __shared__ float sh[DH];
  int i = blockIdx.x, t = threadIdx.x;
  sh[t] = g[i * DH + t] * Ax[i * DH + t]; __syncthreads();
  for (int o = DH / 2; o > 0; o >>= 1) { if (t < o) sh[t] += sh[t + o]; __syncthreads(); }
  if (t == 0) En[i] = 0.5f * (degO[i] * sq[i] + Asq[i] - 2.0f * sh[0]);
}

// scal: [0]=Emax, [1]=Z, [2]=logZ, [3]=S
__global__ __launch_bounds__(1024)
void max_kernel(const float* __restrict__ En, int N, float* __restrict__ scal) {
  __shared__ float sh[1024];
  int t = threadIdx.x;
  float m = -1e30f;
  for (int i = t; i < N; i += 1024) m = fmaxf(m, En[i]);
  sh[t] = m; __syncthreads();
  for (int o = 512; o > 0; o >>= 1) { if (t < o) sh[t] = fmaxf(sh[t], sh[t + o]); __syncthreads(); }
  if (t == 0) scal[0] = sh[0];
}

__global__ __launch_bounds__(1024)
void zu_kernel(const float* __restrict__ En, int N, float* __restrict__ scal) {
  __shared__ float shZ[1024];
  __shared__ float shU[1024];
  int t = threadIdx.x;
  float Emax = scal[0] + 1e-12f;
  float Z = 0.0f, U = 0.0f;
  for (int i = t; i < N; i += 1024) {
    float l = -(En[i] / Emax) / TEMP_F;   // in [-1/T, ~0]: exp is safe unshifted
    float e = expf(l);
    Z += e; U += e * l;
  }
  shZ[t] = Z; shU[t] = U; __syncthreads();
  for (int o = 512; o > 0; o >>= 1) {
    if (t < o) { shZ[t] += shZ[t + o]; shU[t] += shU[t + o]; }
    __syncthreads();
  }
  if (t == 0) {
    float Zt = shZ[0], logZ = logf(Zt);
    scal[1] = Zt; scal[2] = logZ;
    scal[3] = logZ - shU[0] / Zt;         // S = -sum P logP
  }
}

__global__ void c_kernel(float* __restrict__ c, const float* __restrict__ En,
                         const float* __restrict__ scal, int N) {
  int i = blockIdx.x * blockDim.x + threadIdx.x;
  if (i < N) {
    float Emax = scal[0] + 1e-12f, Z = scal[1], logZ = scal[2], S = scal[3];
    float l = -(En[i] / Emax) / TEMP_F;
    float P = expf(l) / Z;
    float logP = l - logZ;
    c[i] = P * (logP + S) / TEMP_F;
  }
}

// ATc[dst] += c[src]; ATcx[dst] += c[src]*g[src]
__global__ void atc_scatter_kernel(float* __restrict__ ATc, float* __restrict__ ATcx,
                                   const float* __restrict__ c, const float* __restrict__ g,
                                   const int* __restrict__ src, const int* __restrict__ dst, int E) {
  int t  = threadIdx.x;
  int e0 = blockIdx.x * EPB;
  int e1 = min(e0 + EPB, E);
  for (int e = e0; e < e1; ++e) {
    int s = src[e], d = dst[e];
    float cs = c[s];
    atomicAdd(&ATcx[d * DH + t], cs * g[s * DH + t]);
    if (t == 0) atomicAdd(&ATc[d], cs);
  }
}

// eg = c*(degO*g - Ax) + ATc*g - ATcx ; t = relu(g + eg) ; LayerNorm -> xcur
__global__ __launch_bounds__(DH)
void finish_kernel(float* __restrict__ xcur, const float* __restrict__ g,
                   const float* __restrict__ Ax, const float* __restrict__ ATcx,
                   const float* __restrict__ c, const float* __restrict__ ATc,
                   const float* __restrict__ degO, const float* __restrict__ gamma,
                   const float* __restrict__ beta) {
  __shared__ float sh[DH];
  int i = blockIdx.x, t = threadIdx.x;
  float gv = g[i * DH + t];
  float eg = c[i] * (degO[i] * gv - Ax[i * DH + t]) + ATc[i] * gv - ATcx[i * DH + t];
  float tv = fmaxf(gv + eg, 0.0f);
  sh[t] = tv; __syncthreads();
  for (int o = DH / 2; o > 0; o >>= 1) { if (t < o) sh[t] += sh[t + o]; __syncthreads(); }
  float mu = sh[0] * (1.0f / DH);
  __syncthreads();
  float dv = tv - mu;
  sh[t] = dv * dv; __syncthreads();
  for (int o = DH / 2; o > 0; o >>= 1) { if (t < o) sh[t] += sh[t + o]; __syncthreads(); }
  float var = sh[0] * (1.0f / DH);
  xcur[i * DH + t] = dv * rsqrtf(var + LN_EPS) * gamma[t] + beta[t];
}

// ---------------------------------------------------------------------------
extern "C" void kernel_launch(void* const* d_in, const int* in_sizes, int n_in,
                              void* d_out, int out_size, void* d_ws, size_t ws_size,
                              hipStream_t stream) {
  (void)n_in; (void)out_size; (void)ws_size;
  const float* x    = (const float*)d_in[0];
  const int*   ei   = (const int*)d_in[1];
  const float* W1   = (const float*)d_in[2];
  const float* b1   = (const float*)d_in[3];
  const float* W2   = (const float*)d_in[4];
  const float* b2   = (const float*)d_in[5];
  const float* Wout = (const float*)d_in[6];
  const float* bout = (const float*)d_in[7];
  const float* gam  = (const float*)d_in[8];
  const float* bet  = (const float*)d_in[9];

  const int N    = in_sizes[0] / DH;   // 8000
  const int E    = in_sizes[1] / 2;    // 256000
  const int DOUT = in_sizes[7];        // 64
  const int* src = ei;
  const int* dst = ei + E;

  float* ws = (float*)d_ws;
  float* xcur = ws;  ws += (size_t)N * DH;
  float* hbuf = ws;  ws += (size_t)N * DH;
  float* gbuf = ws;  ws += (size_t)N * DH;
  float* Ax   = ws;  ws += (size_t)N * DH;
  float* ATcx = ws;  ws += (size_t)N * DH;
  float* degIn= ws;  ws += N;
  float* degO = ws;  ws += N;
  float* dinv = ws;  ws += N;
  float* sq   = ws;  ws += N;
  float* Asq  = ws;  ws += N;
  float* En   = ws;  ws += N;
  float* cbuf = ws;  ws += N;
  float* ATc  = ws;  ws += N;
  float* scal = ws;  ws += 8;

  const int egrid = (E + EPB - 1) / EPB;

  // Degrees (edge structure is layer-invariant)
  hipMemsetAsync(degIn, 0, (size_t)N * sizeof(float), stream);
  hipMemsetAsync(degO,  0, (size_t)N * sizeof(float), stream);
  degree_kernel<<<(E + 255) / 256, 256, 0, stream>>>(src, dst, E, degIn, degO);
  dinv_kernel<<<(N + 255) / 256, 256, 0, stream>>>(degIn, dinv, N);

  const float* Xin = x;
  const float* Wl[2] = {W1, W2};
  const float* bl[2] = {b1, b2};
  for (int layer = 0; layer < 2; ++layer) {
    gemm_wmma_f32<<<dim3(N / 16, DH / 16), 32, 0, stream>>>(Xin, Wl[layer], hbuf, N, DH, DH);
    agg_init_kernel<<<(N * DH + 255) / 256, 256, 0, stream>>>(gbuf, hbuf, dinv, bl[layer], N, DH);
    agg_edge_kernel<<<egrid, DH, 0, stream>>>(gbuf, hbuf, src, dst, dinv, E, DH);

    hipMemsetAsync(Ax,   0, (size_t)N * DH * sizeof(float), stream);
    hipMemsetAsync(ATcx, 0, (size_t)N * DH * sizeof(float), stream);
    hipMemsetAsync(Asq,  0, (size_t)N * sizeof(float), stream);
    hipMemsetAsync(ATc,  0, (size_t)N * sizeof(float), stream);

    node_sq_kernel<<<N, DH, 0, stream>>>(sq, gbuf);
    ax_scatter_kernel<<<egrid, DH, 0, stream>>>(Ax, Asq, gbuf, sq, src, dst, E);
    en_kernel<<<N, DH, 0, stream>>>(En, gbuf, Ax, sq, Asq, degO);
    max_kernel<<<1, 1024, 0, stream>>>(En, N, scal);
    zu_kernel<<<1, 1024, 0, stream>>>(En, N, scal);
    c_kernel<<<(N + 255) / 256, 256, 0, stream>>>(cbuf, En, scal, N);
    atc_scatter_kernel<<<egrid, DH, 0, stream>>>(ATc, ATcx, cbuf, gbuf, src, dst, E);
    finish_kernel<<<N, DH, 0, stream>>>(xcur, gbuf, Ax, ATcx, cbuf, ATc, degO, gam, bet);
    Xin = xcur;
  }

  // Output layer: embedding = gcn_conv(xcur, W_out, b_out) -> d_out [N, DOUT]
  float* out = (float*)d_out;
  gemm_wmma_f32<<<dim3(N / 16, DOUT / 16), 32, 0, stream>>>(xcur, Wout, hbuf, N, DH, DOUT);
  agg_init_kernel<<<(N * DOUT + 255) / 256, 256, 0, stream>>>(out, hbuf, dinv, bout, N, DOUT);
  agg_edge_kernel<<<egrid, DOUT, 0, stream>>>(out, hbuf, src, dst, dinv, E, DOUT);
}